// DistTriangHead_6794638262655
// MI455X (gfx1250) — compile-verified
//
#include <hip/hip_runtime.h>
#include <hip/hip_bf16.h>

typedef float v2f __attribute__((ext_vector_type(2)));
typedef float v8f __attribute__((ext_vector_type(8)));

#define JNT 20
#define NB  16
#define NV  6
#define HW  4096   // 64*64 per map

// d_out offsets (floats)
#define OFF_KP3D   0       // 16*20*3 = 960
#define OFF_RES    960     // 16*20   = 320
#define OFF_CROP   1280    // 96*20*3 = 5760
#define OFF_REPROJ 7040    // 16*6*20*2 = 3840
#define OFF_HM2D   10880   // 96*20*3 = 5760

// ---------------------------------------------------------------------------
// Kernel 1: one block per (bc,j) map. Streams the 4096-elem heatmap once
// (cached in registers), computes softmax stats, soft-argmax coords,
// confidence, and the 4 conf^2 moments that fully determine G.
// ---------------------------------------------------------------------------
__global__ __launch_bounds__(256)
void kp_moments_kernel(const float* __restrict__ heat,
                       float* __restrict__ out,
                       float* __restrict__ moments)
{
    __shared__ float red[256];
    const int m = blockIdx.x;          // map index = bc*20 + j  (bc = b*6+v)
    const int t = threadIdx.x;
    const float* base = heat + (size_t)m * HW;

    // single global read, cached in registers; prefetch hint (global_prefetch_b8)
    float z[16];
#pragma unroll
    for (int i = 0; i < 16; ++i) {
        __builtin_prefetch(base + t + i * 256, 0, 0);
        z[i] = 100.0f * base[t + i * 256];
    }

    // ---- block max ----
    float mx = z[0];
#pragma unroll
    for (int i = 1; i < 16; ++i) mx = fmaxf(mx, z[i]);
    red[t] = mx; __syncthreads();
    for (int s = 128; s > 0; s >>= 1) {
        if (t < s) red[t] = fmaxf(red[t], red[t + s]);
        __syncthreads();
    }
    const float zmax = red[0];
    __syncthreads();

    auto bsum = [&](float v) -> float {
        red[t] = v; __syncthreads();
        for (int s = 128; s > 0; s >>= 1) {
            if (t < s) red[t] += red[t + s];
            __syncthreads();
        }
        float r = red[0]; __syncthreads();
        return r;
    };

    // ---- exp + weighted sums ----
    float e[16];
    float sE = 0.f, sEx = 0.f, sEy = 0.f;
#pragma unroll
    for (int i = 0; i < 16; ++i) {
        const int idx = t + i * 256;
        const float ev = expf(z[i] - zmax);
        e[i] = ev;
        const float col = (float)(idx & 63);
        const float row = (float)(idx >> 6);
        sE  += ev;
        sEx += ev * col;
        sEy += ev * row;
    }
    const float S1   = bsum(sE);
    const float SEx  = bsum(sEx);
    const float SEy  = bsum(sEy);
    const float inv  = 1.0f / S1;
    const float xk   = SEx * inv;
    const float yk   = SEy * inv;
    const float conf = inv;            // max(softmax)=exp(0)/S1, sum(softmax)~1

    if (t == 0) {
        out[OFF_HM2D + m * 3 + 0] = xk;
        out[OFF_HM2D + m * 3 + 1] = yk;
        out[OFF_HM2D + m * 3 + 2] = conf;
        out[OFF_CROP + m * 3 + 0] = xk * 4.0f;
        out[OFF_CROP + m * 3 + 1] = yk * 4.0f;
        out[OFF_CROP + m * 3 + 2] = conf;
    }

    // ---- conf^2 moments: clip(h,0.2,1)>0.25 -> c ; u0=4*col, u1=4*row ----
    float S0 = 0.f, Su = 0.f, Sv = 0.f, Sq = 0.f;
#pragma unroll
    for (int i = 0; i < 16; ++i) {
        const float hm = e[i] * inv;
        if (hm > 0.25f) {
            const float c  = fminf(hm, 1.0f);
            const float c2 = c * c;
            const int idx  = t + i * 256;
            const float u0 = 4.0f * (float)(idx & 63);   // grid x = col*4
            const float u1 = 4.0f * (float)(idx >> 6);   // grid y = row*4
            S0 += c2;
            Su += c2 * u0;
            Sv += c2 * u1;
            Sq += c2 * (u0 * u0 + u1 * u1);
        }
    }
    S0 = bsum(S0); Su = bsum(Su); Sv = bsum(Sv); Sq = bsum(Sq);
    if (t == 0) {
        moments[m * 4 + 0] = S0;
        moments[m * 4 + 1] = Su;
        moments[m * 4 + 2] = Sv;
        moments[m * 4 + 3] = Sq;
    }
}

// ---------------------------------------------------------------------------
// Kernel 2: one wave per (b,j). G = P_big^T (18x4)^T * Q (18x4) via
// V_WMMA_F32_16X16X4_F32 (K padded to 20, 5 accumulating WMMAs), then a
// 4x4 Jacobi eigensolve for the smallest eigenpair + reprojection.
// LDS tiles are padded to 16 columns (zeros) so all 32 lanes load
// unconditionally: no EXEC predication, loads pipeline ahead of the WMMAs.
// ---------------------------------------------------------------------------
__global__ __launch_bounds__(32)
void triang_kernel(const float* __restrict__ proj,
                   const float* __restrict__ moments,
                   float* __restrict__ out)
{
    __shared__ float Pm[20][16];  // stacked P rows (v*3+r); cols>=4 and rows>=18 zero
    __shared__ float Qm[20][16];  // Q = W * P per view; padded with zeros
    __shared__ float Gs[4][4];

    const int bj = blockIdx.x;
    const int b  = bj / JNT;
    const int j  = bj % JNT;
    const int L  = threadIdx.x;

    // zero-fill both tiles (320 floats each, 10 per lane)
    for (int idx = L; idx < 20 * 16; idx += 32) {
        Pm[idx >> 4][idx & 15] = 0.0f;
        Qm[idx >> 4][idx & 15] = 0.0f;
    }
    __syncthreads();

    // load proj[b] (6 views * 3 rows * 4 = 72 floats) into Pm cols 0..3
    for (int idx = L; idx < 72; idx += 32) {
        Pm[idx >> 2][idx & 3] = proj[b * 72 + idx];
    }
    __syncthreads();

    // build Q rows (18 real rows, cols 0..3)
    if (L < 18) {
        const int v = L / 3, r = L % 3;
        const float* Mv = moments + ((size_t)((b * NV + v) * JNT + j)) * 4;
        const float S0 = Mv[0], Su = Mv[1], Sv = Mv[2], Sq = Mv[3];
#pragma unroll
        for (int a = 0; a < 4; ++a) {
            const float p0 = Pm[3 * v + 0][a];
            const float p1 = Pm[3 * v + 1][a];
            const float p2 = Pm[3 * v + 2][a];
            float q;
            if      (r == 0) q =  S0 * p0 - Su * p2;
            else if (r == 1) q =  S0 * p1 - Sv * p2;
            else             q = -Su * p0 - Sv * p1 + Sq * p2;
            Qm[L][a] = q;
        }
    }
    __syncthreads();

    // WMMA accumulate: D = A(16x4) x B(4x16) + C, looped over K chunks.
    // A layout: lanes 0-15 M=lane {K=k,k+1}; lanes 16-31 M=lane-16 {K=k+2,k+3}
    // B layout: VGPR0 K=k(+2 for hi lanes), N=lane&15 ; VGPR1 K=k+1(+2)
    const int mrow = L & 15;
    const int hi   = L >> 4;

    v2f Aop[5], Bop[5];
#pragma unroll
    for (int c = 0; c < 5; ++c) {
        const int k0 = 4 * c + 2 * hi;
        Aop[c].x = Pm[k0][mrow];          // A[m,k] = P_big[k][m]
        Aop[c].y = Pm[k0 + 1][mrow];
        Bop[c].x = Qm[k0][mrow];          // B[k,n] = Q[k][n]
        Bop[c].y = Qm[k0 + 1][mrow];
    }

    v8f acc = {};
#pragma unroll
    for (int c = 0; c < 5; ++c) {
        acc = __builtin_amdgcn_wmma_f32_16x16x4_f32(
            false, Aop[c], false, Bop[c], (short)0, acc, false, false);
    }

    // D VGPR g: M = g + 8*hi, N = lane&15 -> G[a][n] in lanes 0..3
    if (hi == 0 && mrow < 4) {
        Gs[0][mrow] = acc[0];
        Gs[1][mrow] = acc[1];
        Gs[2][mrow] = acc[2];
        Gs[3][mrow] = acc[3];
    }
    __syncthreads();

    if (L == 0) {
        float G[4][4], V[4][4];
#pragma unroll
        for (int a = 0; a < 4; ++a)
#pragma unroll
            for (int c = 0; c < 4; ++c) {
                G[a][c] = 0.5f * (Gs[a][c] + Gs[c][a]);   // symmetrize
                V[a][c] = (a == c) ? 1.0f : 0.0f;
            }
        // cyclic Jacobi
        for (int sweep = 0; sweep < 12; ++sweep) {
            for (int p = 0; p < 3; ++p) {
                for (int q = p + 1; q < 4; ++q) {
                    const float apq = G[p][q];
                    if (fabsf(apq) < 1e-30f) continue;
                    const float theta = (G[q][q] - G[p][p]) / (2.0f * apq);
                    float tt = 1.0f / (fabsf(theta) + sqrtf(theta * theta + 1.0f));
                    if (theta < 0.0f) tt = -tt;
                    const float c = 1.0f / sqrtf(tt * tt + 1.0f);
                    const float s = tt * c;
#pragma unroll
                    for (int k = 0; k < 4; ++k) {        // G <- G*J
                        const float gkp = G[k][p], gkq = G[k][q];
                        G[k][p] = c * gkp - s * gkq;
                        G[k][q] = s * gkp + c * gkq;
                    }
#pragma unroll
                    for (int k = 0; k < 4; ++k) {        // G <- J^T*G
                        const float gpk = G[p][k], gqk = G[q][k];
                        G[p][k] = c * gpk - s * gqk;
                        G[q][k] = s * gpk + c * gqk;
                    }
#pragma unroll
                    for (int k = 0; k < 4; ++k) {        // V <- V*J
                        const float vkp = V[k][p], vkq = V[k][q];
                        V[k][p] = c * vkp - s * vkq;
                        V[k][q] = s * vkp + c * vkq;
                    }
                }
            }
        }
        int idx = 0;
        float wmin = G[0][0];
#pragma unroll
        for (int k = 1; k < 4; ++k)
            if (G[k][k] < wmin) { wmin = G[k][k]; idx = k; }

        const float w3 = V[3][idx];
        const float px = V[0][idx] / w3;
        const float py = V[1][idx] / w3;
        const float pz = V[2][idx] / w3;

        out[OFF_KP3D + (b * JNT + j) * 3 + 0] = px;
        out[OFF_KP3D + (b * JNT + j) * 3 + 1] = py;
        out[OFF_KP3D + (b * JNT + j) * 3 + 2] = pz;
        out[OFF_RES  + (b * JNT + j)]         = sqrtf(fmaxf(wmin, 1e-12f));

        // reprojection: p = proj[b,c] * [kp,1]; out = p[:2]/p[2]
        for (int c = 0; c < NV; ++c) {
            const float* Pc = proj + (size_t)(b * NV + c) * 12;
            const float p0 = Pc[0] * px + Pc[1] * py + Pc[2]  * pz + Pc[3];
            const float p1 = Pc[4] * px + Pc[5] * py + Pc[6]  * pz + Pc[7];
            const float p2 = Pc[8] * px + Pc[9] * py + Pc[10] * pz + Pc[11];
            out[OFF_REPROJ + ((b * NV + c) * JNT + j) * 2 + 0] = p0 / p2;
            out[OFF_REPROJ + ((b * NV + c) * JNT + j) * 2 + 1] = p1 / p2;
        }
    }
}

extern "C" void kernel_launch(void* const* d_in, const int* in_sizes, int n_in,
                              void* d_out, int out_size, void* d_ws, size_t ws_size,
                              hipStream_t stream) {
    const float* heat = (const float*)d_in[0];   // (96,20,64,64) f32
    const float* proj = (const float*)d_in[1];   // (16,6,3,4) f32
    float* out     = (float*)d_out;              // 16640 f32
    float* moments = (float*)d_ws;               // 1920*4 f32 scratch

    kp_moments_kernel<<<96 * JNT, 256, 0, stream>>>(heat, out, moments);
    triang_kernel<<<NB * JNT, 32, 0, stream>>>(proj, moments, out);
}